// RecursiveAttention_73675868995912
// MI455X (gfx1250) — compile-verified
//
#include <hip/hip_runtime.h>
#include <math.h>

typedef float v2f __attribute__((ext_vector_type(2)));
typedef float v8f __attribute__((ext_vector_type(8)));

#define N_Q   4096
#define N_K   32768
#define D_IN  256
#define D_QK  64
#define D_V   64

// workspace layout (in floats)
#define OFF_Q 0
#define OFF_K (OFF_Q + N_Q * D_QK)          // 262144
#define OFF_V (OFF_K + N_K * D_QK)          // 2359296
#define OFF_S (OFF_V + N_K * D_V)           // 4456448
#define OFF_M (OFF_S + N_K)                 // 4489216 (uint-encoded row max)
#define OFF_D (OFF_M + N_Q)                 // 4493312 (softmax denom)
#define OFF_C (OFF_D + N_Q)                 // 4497408 (C accumulator)
#define ZERO_DWORDS (N_Q + N_Q + N_Q * D_V) // m_enc + denom + Cacc = 270336

// order-preserving float->uint encoding (monotone under unsigned compare)
__device__ __forceinline__ unsigned enc_f(float x) {
    unsigned b = __float_as_uint(x);
    return (b & 0x80000000u) ? ~b : (b | 0x80000000u);
}
__device__ __forceinline__ float dec_f(unsigned e) {
    return __uint_as_float((e & 0x80000000u) ? (e & 0x7FFFFFFFu) : ~e);
}

// ---------------------------------------------------------------------------
// zero init of m_enc / denom / Cacc region
// ---------------------------------------------------------------------------
__global__ void init_ws_kernel(unsigned* __restrict__ p) {
    int idx = blockIdx.x * blockDim.x + threadIdx.x;
    if (idx < ZERO_DWORDS) p[idx] = 0u;
}

// ---------------------------------------------------------------------------
// Out[M x 64] = X[M x 256] @ W[256 x 64] + bias, fp32 WMMA 16x16x4.
// Block = 128 threads (4 waves); block covers 32 rows. Wave w owns N-tile
// cols [16w, 16w+16) and TWO 16-row M-tiles with independent accumulator
// chains sharing one B fragment per K-step (halves W loads, doubles WMMA ILP
// so the two v_wmma dependency chains interleave and hide XDL latency).
// A-fragment layout (ISA 16x4 f32): lanes 0-15 rows, VGPR pair holds
// K = {0,1} (lane half 0) or {2,3} (lane half 1). B mirrored per-lane-N.
// ---------------------------------------------------------------------------
__global__ void gemm_proj_kernel(const float* __restrict__ X,
                                 const float* __restrict__ W,
                                 const float* __restrict__ bias,
                                 float* __restrict__ Out) {
    const int lane    = threadIdx.x & 31;
    const int wave    = threadIdx.x >> 5;
    const int rowBase = blockIdx.x * 32;
    const int cb      = wave * 16;      // N-tile base (4 waves cover N=64)
    const int ln      = lane & 15;      // row (A) / col (B,C) within tile
    const int h       = lane >> 4;      // lane-half selects K pair

    const float* xrow0 = X + (size_t)(rowBase + ln) * D_IN + 2 * h;
    const float* xrow1 = xrow0 + (size_t)16 * D_IN;

    v8f acc0 = {};
    v8f acc1 = {};
#pragma unroll 4
    for (int k4 = 0; k4 < D_IN; k4 += 4) {
        v2f a0 = *(const v2f*)(xrow0 + k4);
        v2f a1 = *(const v2f*)(xrow1 + k4);
        const float* wp = W + (size_t)(k4 + 2 * h) * D_QK + cb + ln;
        v2f b;
        b.x = wp[0];
        b.y = wp[D_QK];
        acc0 = __builtin_amdgcn_wmma_f32_16x16x4_f32(
            false, a0, false, b, (short)0, acc0, false, false);
        acc1 = __builtin_amdgcn_wmma_f32_16x16x4_f32(
            false, a1, false, b, (short)0, acc1, false, false);
    }

    const float bv = bias[cb + ln];
    // C layout: VGPR r -> row (tileBase + r + 8*h), col (cb + ln)
    float* out0 = Out + (size_t)(rowBase + 8 * h) * D_QK + cb + ln;
    float* out1 = out0 + (size_t)16 * D_QK;
#pragma unroll
    for (int r = 0; r < 8; ++r) {
        out0[(size_t)r * D_QK] = acc0[r] + bv;
        out1[(size_t)r * D_QK] = acc1[r] + bv;
    }
}

// ---------------------------------------------------------------------------
// Pass A: per valid key j, s_j = (Q[row_map[j]] . K_j) / 8; atomic row max.
// Wave per key; lane-parallel dot over 64 dims (2 per lane), shfl_xor reduce.
// ---------------------------------------------------------------------------
__global__ void scores_kernel(const float* __restrict__ Q,
                              const float* __restrict__ K,
                              const int* __restrict__ row_map,
                              float* __restrict__ S,
                              unsigned* __restrict__ Menc) {
    const int lane = threadIdx.x & 31;
    const int key  = blockIdx.x * 8 + (threadIdx.x >> 5);
    if (key >= N_K) return;
    const int rm = row_map[key];
    if (rm < 0) return;

    v2f q = *(const v2f*)(Q + (size_t)rm * D_QK + 2 * lane);
    v2f k = *(const v2f*)(K + (size_t)key * D_QK + 2 * lane);
    float p = q.x * k.x + q.y * k.y;
#pragma unroll
    for (int off = 16; off > 0; off >>= 1)
        p += __shfl_xor(p, off, 32);
    const float s = p * 0.125f; // 1/sqrt(64)

    if (lane == 0) {
        S[key] = s;
        atomicMax(&Menc[rm], enc_f(s));
    }
}

// ---------------------------------------------------------------------------
// Pass B: w = exp(s - m); denom[i] += w; Cacc[i][:] += w * V[j][:]
// ---------------------------------------------------------------------------
__global__ void accum_kernel(const float* __restrict__ S,
                             const float* __restrict__ V,
                             const int* __restrict__ row_map,
                             const unsigned* __restrict__ Menc,
                             float* __restrict__ Den,
                             float* __restrict__ Cacc) {
    const int lane = threadIdx.x & 31;
    const int key  = blockIdx.x * 8 + (threadIdx.x >> 5);
    if (key >= N_K) return;
    const int rm = row_map[key];
    if (rm < 0) return;

    const float m = dec_f(Menc[rm]);
    const float w = expf(S[key] - m);

    if (lane == 0) atomicAdd(&Den[rm], w);

    v2f v = *(const v2f*)(V + (size_t)key * D_V + 2 * lane);
    float* c = Cacc + (size_t)rm * D_V + 2 * lane;
    atomicAdd(c + 0, w * v.x);
    atomicAdd(c + 1, w * v.y);
}

// ---------------------------------------------------------------------------
// Pass C: out = denom > 0 ? Cacc / denom : 0   (denom>0 <=> cnt>0)
// ---------------------------------------------------------------------------
__global__ void normalize_kernel(const float* __restrict__ Cacc,
                                 const float* __restrict__ Den,
                                 float* __restrict__ out) {
    int idx = blockIdx.x * blockDim.x + threadIdx.x;
    if (idx >= N_Q * D_V) return;
    const float d = Den[idx / D_V];
    out[idx] = (d > 0.0f) ? (Cacc[idx] / d) : 0.0f;
}

extern "C" void kernel_launch(void* const* d_in, const int* in_sizes, int n_in,
                              void* d_out, int out_size, void* d_ws, size_t ws_size,
                              hipStream_t stream) {
    const float* X1 = (const float*)d_in[0];
    const float* X2 = (const float*)d_in[1];
    const float* Wq = (const float*)d_in[2];
    const float* bq = (const float*)d_in[3];
    const float* Wk = (const float*)d_in[4];
    const float* bk = (const float*)d_in[5];
    const float* Wv = (const float*)d_in[6];
    const float* bv = (const float*)d_in[7];
    const int* row_map = (const int*)d_in[8];

    float* ws = (float*)d_ws;
    float*    Q    = ws + OFF_Q;
    float*    Kp   = ws + OFF_K;
    float*    Vp   = ws + OFF_V;
    float*    S    = ws + OFF_S;
    unsigned* Menc = (unsigned*)(ws + OFF_M);
    float*    Den  = ws + OFF_D;
    float*    Cacc = ws + OFF_C;

    init_ws_kernel<<<(ZERO_DWORDS + 255) / 256, 256, 0, stream>>>(Menc);

    gemm_proj_kernel<<<N_Q / 32, 128, 0, stream>>>(X1, Wq, bq, Q);
    gemm_proj_kernel<<<N_K / 32, 128, 0, stream>>>(X2, Wk, bk, Kp);
    gemm_proj_kernel<<<N_K / 32, 128, 0, stream>>>(X2, Wv, bv, Vp);

    scores_kernel<<<N_K / 8, 256, 0, stream>>>(Q, Kp, row_map, S, Menc);
    accum_kernel<<<N_K / 8, 256, 0, stream>>>(S, Vp, row_map, Menc, Den, Cacc);
    normalize_kernel<<<(N_Q * D_V + 255) / 256, 256, 0, stream>>>(Cacc, Den, (float*)d_out);
}